// ContrastiveLoss_84619445666637
// MI455X (gfx1250) — compile-verified
//
#include <hip/hip_runtime.h>

// ContrastiveLoss on MI455X (gfx1250).
// Gram matrices via V_WMMA_F32_16X16X4_F32 with 4x4 register blocking
// (64x64 macro-tile per wave -> 0.5 fragment loads per WMMA).
// outputs: [32, 1024, 1024] f32, encode_sentences: [1024, 1024] f32 -> 3 f32.

typedef float v2f __attribute__((ext_vector_type(2)));
typedef float v4f __attribute__((ext_vector_type(4)));
typedef float v8f __attribute__((ext_vector_type(8)));

#define N_SEC  32
#define BATCH  1024
#define DIM    1024
#define MARGIN 1.0f
#define EPS_F  1e-12f

// Workspace layout (requires ~4.33 MB):
//   [0,16)        : 2 doubles (sent_sum, secret_sum)
//   [256, +135168): 33792 floats row norms: outputs rows [0,32768), enc [32768,33792)
//   [135424, +4MB): d_sent matrix 1024x1024 f32
#define WS_NORMS_OFF 256
#define WS_DSENT_OFF 135424

// ---- WMMA fragment loader -------------------------------------------------
// For C = X * Y^T (Gram), A fragment (16x4 rows of X) and B fragment
// (4x16 = rows of Y) share the per-lane pattern:
//   row = lane & 15 ; k = k0 + 2*(lane>>4) .. +1   (one 8-byte load)
__device__ __forceinline__ v2f load_frag(const float* __restrict__ rows,
                                         int ldr, int k0, int lane) {
  const int r  = lane & 15;
  const int kk = k0 + ((lane >> 4) << 1);
  return *(const v2f*)(rows + (size_t)r * (size_t)ldr + kk);
}

__device__ __forceinline__ float wave_reduce(float v) {
  #pragma unroll
  for (int off = 16; off > 0; off >>= 1) v += __shfl_down(v, off, 32);
  return v;
}

// ---- 4x4-blocked Gram: 64x64 macro-tile per wave --------------------------
// 8 loads -> 16 WMMAs per K-step; 16 independent WMMAs also hide the
// WMMA->WMMA accumulator RAW hazard window.
template <int LD>
__device__ __forceinline__ void gram_4x4(const float* __restrict__ rowsA,
                                         const float* __restrict__ rowsB,
                                         int lane, v8f c[4][4]) {
  const int r     = lane & 15;
  const int kbase = (lane >> 4) << 1;
  const float* baseA = rowsA + (size_t)r * LD + kbase;
  const float* baseB = rowsB + (size_t)r * LD + kbase;
  #pragma unroll 2
  for (int k0 = 0; k0 < DIM; k0 += 4) {
    v2f a[4], b[4];
    #pragma unroll
    for (int s = 0; s < 4; ++s)
      a[s] = *(const v2f*)(baseA + (size_t)(s * 16) * LD + k0);
    #pragma unroll
    for (int s = 0; s < 4; ++s)
      b[s] = *(const v2f*)(baseB + (size_t)(s * 16) * LD + k0);
    #pragma unroll
    for (int si = 0; si < 4; ++si)
      #pragma unroll
      for (int sj = 0; sj < 4; ++sj)
        c[si][sj] = __builtin_amdgcn_wmma_f32_16x16x4_f32(
            false, a[si], false, b[sj], (short)0, c[si][sj], false, false);
  }
}

// ---- Kernel 0: zero accumulators ------------------------------------------
__global__ void init_acc_kernel(double* __restrict__ acc) {
  if (threadIdx.x < 2) acc[threadIdx.x] = 0.0;
}

// ---- Kernel 1: row norms (||x||^2) for all 33*1024 rows -------------------
__global__ __launch_bounds__(128)
void norms_kernel(const float* __restrict__ outputs,
                  const float* __restrict__ enc,
                  float* __restrict__ norms) {
  const int wave = (blockIdx.x * blockDim.x + threadIdx.x) >> 5;
  const int lane = threadIdx.x & 31;
  if (wave >= (N_SEC + 1) * BATCH) return;
  const float* row = (wave < N_SEC * BATCH)
                         ? outputs + (size_t)wave * DIM
                         : enc + (size_t)(wave - N_SEC * BATCH) * DIM;
  float s = 0.f;
  #pragma unroll
  for (int it = 0; it < DIM / 128; ++it) {
    v4f x = *(const v4f*)(row + it * 128 + lane * 4);
    s += x.x * x.x + x.y * x.y + x.z * x.z + x.w * x.w;
  }
  s = wave_reduce(s);
  if (lane == 0) norms[wave] = s;
}

// ---- Kernel 2: d_sent via blocked Gram; one wave per 64x64 macro-tile -----
__global__ __launch_bounds__(128)
void dsent_kernel(const float* __restrict__ enc,
                  const float* __restrict__ norms,
                  float* __restrict__ dsent) {
  const int wave = (blockIdx.x * blockDim.x + threadIdx.x) >> 5;
  const int lane = threadIdx.x & 31;
  const int TM = wave >> 4, TN = wave & 15;   // 16x16 macro grid

  v8f c[4][4] = {};
  gram_4x4<DIM>(enc + (size_t)(TM * 64) * DIM,
                enc + (size_t)(TN * 64) * DIM, lane, c);

  const float* nA = norms + N_SEC * BATCH + TM * 64;
  const float* nB = norms + N_SEC * BATCH + TN * 64;
  const int h = lane >> 4, n = lane & 15;
  #pragma unroll
  for (int sj = 0; sj < 4; ++sj) {
    const float sqn = nB[sj * 16 + n];
    #pragma unroll
    for (int si = 0; si < 4; ++si) {
      #pragma unroll
      for (int v = 0; v < 8; ++v) {
        const int m = si * 16 + v + 8 * h;
        float d2 = fmaxf(nA[m] + sqn - 2.f * c[si][sj][v], 0.f);
        dsent[(size_t)(TM * 64 + m) * BATCH + TN * 64 + sj * 16 + n] =
            sqrtf(d2 + EPS_F);
      }
    }
  }
}

// ---- Kernel 3: sentence loss over upper-triangle 64x64 macro-tiles --------
// wave -> (i, macro pair p in [0,136)); off-diagonal macro-tiles weighted x2.
__global__ __launch_bounds__(128)
void sentence_kernel(const float* __restrict__ outputs,
                     const float* __restrict__ norms,
                     const float* __restrict__ dsent,
                     double* __restrict__ acc) {
  __shared__ float part[4];
  const int wib  = threadIdx.x >> 5;
  const int lane = threadIdx.x & 31;
  const int w = blockIdx.x * 4 + wib;
  const int i = w / 136;
  int rem = w % 136;
  int TM = 0;
  while (rem >= 16 - TM) { rem -= 16 - TM; ++TM; }
  const int TN = TM + rem;

  const float* X = outputs + (size_t)i * BATCH * DIM;
  v8f c[4][4] = {};
  gram_4x4<DIM>(X + (size_t)(TM * 64) * DIM,
                X + (size_t)(TN * 64) * DIM, lane, c);

  const float* nA = norms + i * BATCH + TM * 64;
  const float* nB = norms + i * BATCH + TN * 64;
  const int h = lane >> 4, n = lane & 15;
  float s = 0.f;
  #pragma unroll
  for (int sj = 0; sj < 4; ++sj) {
    const float sqn = nB[sj * 16 + n];
    #pragma unroll
    for (int si = 0; si < 4; ++si) {
      #pragma unroll
      for (int v = 0; v < 8; ++v) {
        const int m = si * 16 + v + 8 * h;
        float d2 = fmaxf(nA[m] + sqn - 2.f * c[si][sj][v], 0.f);
        float d  = sqrtf(d2 + EPS_F);
        float ds = dsent[(size_t)(TM * 64 + m) * BATCH + TN * 64 + sj * 16 + n];
        float df = d - ds;
        s += df * df;
      }
    }
  }
  if (TM != TN) s *= 2.f;
  s = wave_reduce(s);
  if (lane == 0) part[wib] = s;
  __syncthreads();
  if (threadIdx.x == 0) {
    double t = (double)part[0] + (double)part[1] + (double)part[2] + (double)part[3];
    atomicAdd(acc + 0, t);
  }
}

// ---- Kernel 4: secret (margin) loss ---------------------------------------
// Per b: 32x32 Gram of outputs[:, b, :] via 3 upper 16x16 tiles.
__global__ __launch_bounds__(128)
void secret_kernel(const float* __restrict__ outputs,
                   const float* __restrict__ norms,
                   double* __restrict__ acc) {
  __shared__ float part[4];
  const int wib  = threadIdx.x >> 5;
  const int lane = threadIdx.x & 31;
  const int w = blockIdx.x * 4 + wib;
  const int b = w / 3;
  const int t = w % 3;
  const int tm = (t == 2) ? 1 : 0;
  const int tn = (t == 0) ? 0 : 1;

  const float* base = outputs + (size_t)b * DIM;   // secret row n at +n*B*D
  const float* rowsA = base + (size_t)(tm * 16) * (size_t)(BATCH * DIM);
  const float* rowsB = base + (size_t)(tn * 16) * (size_t)(BATCH * DIM);
  v8f c = {};
  #pragma unroll 4
  for (int k0 = 0; k0 < DIM; k0 += 4) {
    v2f a = load_frag(rowsA, BATCH * DIM, k0, lane);
    v2f bb = load_frag(rowsB, BATCH * DIM, k0, lane);
    c = __builtin_amdgcn_wmma_f32_16x16x4_f32(false, a, false, bb,
                                              (short)0, c, false, false);
  }

  const int h = lane >> 4, n = lane & 15;
  const int j = tn * 16 + n;
  const float sqj = norms[(size_t)j * BATCH + b];
  float s = 0.f;
  #pragma unroll
  for (int v = 0; v < 8; ++v) {
    const int ii = tm * 16 + v + 8 * h;
    if (ii < j) {
      const float sqi = norms[(size_t)ii * BATCH + b];
      float d2 = fmaxf(sqi + sqj - 2.f * c[v], 0.f);
      float d  = sqrtf(d2 + EPS_F);
      s += fmaxf(MARGIN - d, 0.f);
    }
  }
  s = wave_reduce(s);
  if (lane == 0) part[wib] = s;
  __syncthreads();
  if (threadIdx.x == 0) {
    double t2 = (double)part[0] + (double)part[1] + (double)part[2] + (double)part[3];
    atomicAdd(acc + 1, t2);
  }
}

// ---- Kernel 5: finalize ----------------------------------------------------
__global__ void finalize_kernel(const double* __restrict__ acc,
                                float* __restrict__ out) {
  if (threadIdx.x == 0) {
    double sent   = acc[0] / ((double)N_SEC * BATCH * BATCH);
    double secret = acc[1] / (double)BATCH / ((double)N_SEC * (N_SEC - 1) / 2.0);
    double loss   = 0.5 * sent + 0.5 * secret;
    out[0] = (float)loss;
    out[1] = (float)sent;
    out[2] = (float)secret;
  }
}

extern "C" void kernel_launch(void* const* d_in, const int* in_sizes, int n_in,
                              void* d_out, int out_size, void* d_ws, size_t ws_size,
                              hipStream_t stream) {
  (void)in_sizes; (void)n_in; (void)out_size; (void)ws_size;
  const float* outputs = (const float*)d_in[0];  // [32,1024,1024]
  const float* enc     = (const float*)d_in[1];  // [1024,1024]
  float* out = (float*)d_out;

  char* ws = (char*)d_ws;
  double* acc   = (double*)ws;
  float*  norms = (float*)(ws + WS_NORMS_OFF);
  float*  dsent = (float*)(ws + WS_DSENT_OFF);

  init_acc_kernel<<<1, 32, 0, stream>>>(acc);
  // 33792 rows, 4 waves/block -> 8448 blocks
  norms_kernel<<<8448, 128, 0, stream>>>(outputs, enc, norms);
  // 256 macro-tiles, 4 waves/block -> 64 blocks
  dsent_kernel<<<64, 128, 0, stream>>>(enc, norms, dsent);
  // 32 * 136 macro pairs = 4352 waves -> 1088 blocks
  sentence_kernel<<<1088, 128, 0, stream>>>(outputs, norms, dsent, acc);
  // 1024 * 3 tiles = 3072 waves -> 768 blocks
  secret_kernel<<<768, 128, 0, stream>>>(outputs, norms, acc);
  finalize_kernel<<<1, 32, 0, stream>>>(acc, out);
}